// SappyEncoderModule_25718264168640
// MI455X (gfx1250) — compile-verified
//
#include <hip/hip_runtime.h>

typedef __attribute__((ext_vector_type(2))) float v2f;
typedef __attribute__((ext_vector_type(8))) float v8f;

#define N_NODES 40000
#define N_EDGES 640000
#define D 128
#define LDS_STRIDE 132   // 128 + 4 pad -> conflict-free column reads

// ---------------------------------------------------------------- zero ----
__global__ void zero_f4_kernel(float4* __restrict__ p, int n4) {
    int i = blockIdx.x * blockDim.x + threadIdx.x;
    if (i < n4) p[i] = make_float4(0.f, 0.f, 0.f, 0.f);
}

// ------------------------------------------------------- edge scatter ----
// One wave32 per edge: lane L handles features [4L, 4L+4).
// msg = x[src] / attr  ;  agg[dst] += msg  (L2-side f32 atomics)
__global__ __launch_bounds__(256)
void edge_scatter_kernel(const float* __restrict__ x,
                         const long long* __restrict__ src,
                         const long long* __restrict__ dst,
                         const float* __restrict__ attr,
                         float* __restrict__ agg) {
    const int lane = threadIdx.x & 31;
    const int e = blockIdx.x * (blockDim.x >> 5) + (threadIdx.x >> 5);
    if (e >= N_EDGES) return;

    const long long s = src[e];
    const long long d = dst[e];
    const float a = attr[e];

    const float4 xv = *(const float4*)(x + (size_t)s * D + lane * 4);
    float4 m;
    m.x = xv.x / a; m.y = xv.y / a; m.z = xv.z / a; m.w = xv.w / a;

    float* o = agg + (size_t)d * D + lane * 4;
    __hip_atomic_fetch_add(o + 0, m.x, __ATOMIC_RELAXED, __HIP_MEMORY_SCOPE_AGENT);
    __hip_atomic_fetch_add(o + 1, m.y, __ATOMIC_RELAXED, __HIP_MEMORY_SCOPE_AGENT);
    __hip_atomic_fetch_add(o + 2, m.z, __ATOMIC_RELAXED, __HIP_MEMORY_SCOPE_AGENT);
    __hip_atomic_fetch_add(o + 3, m.w, __ATOMIC_RELAXED, __HIP_MEMORY_SCOPE_AGENT);
}

// ------------------------------------------------- GEMM (WMMA f32) -------
// out[m,n] = relu( sum_k A[m,k] * W[n,k] + b[n] )
// Block: 256 threads = 8 waves. Block computes 16 rows x 128 cols.
// Wave w computes the 16x16 tile at columns [16w, 16w+16).
// K-loop: V_WMMA_F32_16X16X4_F32, 32 steps of K=4.
//
// f32 A-fragment layout (ISA 7.12.2, 16x4):
//   VGPR0: lanes 0-15 -> A[M=lane][k0],   lanes 16-31 -> A[M=lane-16][k0+2]
//   VGPR1: lanes 0-15 -> A[M=lane][k0+1], lanes 16-31 -> A[M=lane-16][k0+3]
// => per-lane v2f load at (row = lane&15, col = k0 + 2*(lane>>4)).
// B (4x16) mirrors this with B[k][n] = W[n][k]: per-lane v2f load from
// W row (n0 + (lane&15)) at col k0 + 2*(lane>>4).
__global__ __launch_bounds__(256)
void gemm_bias_relu_kernel(const float* __restrict__ A,   // [N_NODES, D]
                           const float* __restrict__ W,   // [D, D]
                           const float* __restrict__ bias,// [D]
                           float* __restrict__ out) {     // [N_NODES, D]
    __shared__ float lA[16 * LDS_STRIDE];

    const int t  = threadIdx.x;
    const int m0 = blockIdx.x * 16;

    // Cooperative load: the 16x128 A tile is one contiguous 8 KB chunk.
    const float4* src4 = (const float4*)(A + (size_t)m0 * D);  // 512 float4
#pragma unroll
    for (int it = 0; it < 2; ++it) {
        int idx4 = t + it * 256;          // 0..511
        int i    = idx4 * 4;              // float index in tile
        int row  = i >> 7;
        int col  = i & 127;
        *(float4*)&lA[row * LDS_STRIDE + col] = src4[idx4];
    }
    __syncthreads();

    const int wave = t >> 5;
    const int lane = t & 31;
    const int n0   = wave * 16;
    const int lr   = lane & 15;
    const int hi   = lane >> 4;

    const float* arow = &lA[lr * LDS_STRIDE + hi * 2];
    const float* wrow = W + (size_t)(n0 + lr) * D + hi * 2;

    v8f acc = {0.f, 0.f, 0.f, 0.f, 0.f, 0.f, 0.f, 0.f};
#pragma unroll
    for (int k0 = 0; k0 < D; k0 += 4) {
        v2f av = *(const v2f*)(arow + k0);
        v2f bv = *(const v2f*)(wrow + k0);
        acc = __builtin_amdgcn_wmma_f32_16x16x4_f32(
            /*neg_a=*/false, av, /*neg_b=*/false, bv,
            /*c_mod=*/(short)0, acc, /*reuse_a=*/false, /*reuse_b=*/false);
    }

    const float bv = bias[n0 + lr];
    // C/D layout: VGPR r -> M = r + 8*hi (lanes 0-15: M=r; 16-31: M=r+8), N = lr
    float* orow = out + (size_t)(m0 + hi * 8) * D + n0 + lr;
#pragma unroll
    for (int r = 0; r < 8; ++r) {
        float y = acc[r] + bv;
        orow[(size_t)r * D] = y > 0.f ? y : 0.f;   // leaky(0.01) then max(,0) == relu
    }
}

// ---------------------------------------------------------------- host ---
extern "C" void kernel_launch(void* const* d_in, const int* in_sizes, int n_in,
                              void* d_out, int out_size, void* d_ws, size_t ws_size,
                              hipStream_t stream) {
    const float*      x    = (const float*)d_in[0];
    const long long*  ei   = (const long long*)d_in[1];   // [2, E] int64
    const float*      attr = (const float*)d_in[2];
    const float*      W1   = (const float*)d_in[3];
    const float*      b1   = (const float*)d_in[4];
    const float*      W2   = (const float*)d_in[5];
    const float*      b2   = (const float*)d_in[6];
    float*            out  = (float*)d_out;

    const long long* src = ei;            // row 0
    const long long* dst = ei + N_EDGES;  // row 1

    float* agg = (float*)d_ws;                       // [N_NODES * D]
    float* h1  = agg + (size_t)N_NODES * D;          // [N_NODES * D]

    const int n4 = N_NODES * D / 4;                  // 1,280,000 float4
    const dim3 zgrid((n4 + 255) / 256), zblk(256);
    const dim3 egrid(N_EDGES / 8), eblk(256);        // 8 edges per block
    const dim3 ggrid(N_NODES / 16), gblk(256);       // 2500 row-tiles

    // ---- layer 1 ----
    zero_f4_kernel<<<zgrid, zblk, 0, stream>>>((float4*)agg, n4);
    edge_scatter_kernel<<<egrid, eblk, 0, stream>>>(x, src, dst, attr, agg);
    gemm_bias_relu_kernel<<<ggrid, gblk, 0, stream>>>(agg, W1, b1, h1);

    // ---- layer 2 ----
    zero_f4_kernel<<<zgrid, zblk, 0, stream>>>((float4*)agg, n4);
    edge_scatter_kernel<<<egrid, eblk, 0, stream>>>(h1, src, dst, attr, agg);
    gemm_bias_relu_kernel<<<ggrid, gblk, 0, stream>>>(agg, W2, b2, out);
}